// Attention_63273458205325
// MI455X (gfx1250) — compile-verified
//
#include <hip/hip_runtime.h>
#include <hip/hip_bf16.h>
#include <math.h>

#define B_      64
#define C_      256
#define HW      4096
#define HW4     1024            // HW in float4 units
#define PLANES  128
#define HIDDEN  16
#define K_      8
#define ATT     4
#define NCHUNK  4               // chunks of 1024 spatial positions
#define CHUNK   1024
#define EPSF    1e-5f
#define TEMPF   30.0f

typedef __attribute__((ext_vector_type(2))) float v2f;
typedef __attribute__((ext_vector_type(8))) float v8f;

__device__ __forceinline__ float waveSum(float v) {
#pragma unroll
    for (int o = 16; o > 0; o >>= 1) v += __shfl_xor(v, o, 32);
    return v;
}
__device__ __forceinline__ float waveMax(float v) {
#pragma unroll
    for (int o = 16; o > 0; o >>= 1) v = fmaxf(v, __shfl_xor(v, o, 32));
    return v;
}

// ---------------------------------------------------------------------------
// Pass 1 over x (bandwidth-bound, float4): per (b, 1024-spatial chunk):
//   - per-channel partial sums (beta_c)  [3-stage butterfly + padded LDS fold]
//   - mask logits (dot over channels with w_mask)
//   - per-spatial channel-mean / channel-max, summed over the chunk (beta_s)
// Thread t owns spatial positions s0+4t..s0+4t+3.
// ---------------------------------------------------------------------------
__global__ void pass1_kernel(const float* __restrict__ x,
                             const float* __restrict__ w_mask,
                             const float* __restrict__ b_mask,
                             float* __restrict__ g_mask,
                             float* __restrict__ g_bc_part,
                             float* __restrict__ g_a_part,
                             float* __restrict__ g_m_part) {
    int b     = blockIdx.x >> 2;
    int chunk = blockIdx.x & 3;
    int tid   = threadIdx.x;
    int wave  = tid >> 5, lane = tid & 31;
    int s0    = chunk * CHUNK;

    const float4* xb4 = (const float4*)(x + (size_t)b * C_ * HW + s0);

    __shared__ float chpart[32 * 257];   // 32 partial slots/channel, bank-padded
    __shared__ float sh[16];

    float m0 = 0.f, m1 = 0.f, m2 = 0.f, m3 = 0.f;       // mask logit accum
    float c0 = 0.f, c1 = 0.f, c2 = 0.f, c3 = 0.f;       // channel sum per spatial
    float x0 = -INFINITY, x1 = -INFINITY, x2 = -INFINITY, x3 = -INFINITY;

#pragma unroll 2
    for (int c = 0; c < C_; ++c) {
        float4 v = xb4[(size_t)c * HW4 + tid];
        if (c + 1 < C_) __builtin_prefetch(&xb4[(size_t)(c + 1) * HW4 + tid], 0, 1);
        float wm = w_mask[c];
        m0 = fmaf(wm, v.x, m0); m1 = fmaf(wm, v.y, m1);
        m2 = fmaf(wm, v.z, m2); m3 = fmaf(wm, v.w, m3);
        c0 += v.x; c1 += v.y; c2 += v.z; c3 += v.w;
        x0 = fmaxf(x0, v.x); x1 = fmaxf(x1, v.y);
        x2 = fmaxf(x2, v.z); x3 = fmaxf(x3, v.w);
        // per-channel spatial sum: 3-stage butterfly -> 4 partial lanes/wave
        float p = (v.x + v.y) + (v.z + v.w);
        p += __shfl_xor(p, 16, 32);
        p += __shfl_xor(p, 8, 32);
        p += __shfl_xor(p, 4, 32);
        if (lane < 4) chpart[(wave * 4 + lane) * 257 + c] = p;
    }
    __syncthreads();
    // fold 32 partial slots per channel; thread t handles channel t
    float cs = 0.f;
#pragma unroll
    for (int i = 0; i < 32; ++i) cs += chpart[i * 257 + tid];
    g_bc_part[(size_t)(b * NCHUNK + chunk) * C_ + tid] = cs;

    float bm = b_mask[0];
    float4 mo;
    mo.x = m0 + bm; mo.y = m1 + bm; mo.z = m2 + bm; mo.w = m3 + bm;
    ((float4*)(g_mask + (size_t)b * HW + s0))[tid] = mo;

    float r1 = waveSum(((c0 + c1) + (c2 + c3)) * (1.0f / 256.0f)); // mean over C
    float r2 = waveSum((x0 + x1) + (x2 + x3));                     // max over C, summed
    if (lane == 0) { sh[wave] = r1; sh[8 + wave] = r2; }
    __syncthreads();
    if (tid == 0) {
        float sa = 0.f, sm = 0.f;
        for (int w = 0; w < 8; ++w) { sa += sh[w]; sm += sh[8 + w]; }
        g_a_part[b * NCHUNK + chunk] = sa;
        g_m_part[b * NCHUNK + chunk] = sm;
    }
}

// ---------------------------------------------------------------------------
// Per-b: fold chunk partials; softmax over the 4096 mask logits (in place).
// ---------------------------------------------------------------------------
__global__ void reduce_softmax_kernel(float* __restrict__ g_mask,
                                      const float* __restrict__ g_bc_part,
                                      const float* __restrict__ g_a_part,
                                      const float* __restrict__ g_m_part,
                                      float* __restrict__ g_betac,
                                      float* __restrict__ g_a,
                                      float* __restrict__ g_m) {
    int b = blockIdx.x, tid = threadIdx.x;
    int wave = tid >> 5, lane = tid & 31;
    __shared__ float sh[8];

    float s = 0.f;
#pragma unroll
    for (int ch = 0; ch < NCHUNK; ++ch)
        s += g_bc_part[(size_t)(b * NCHUNK + ch) * C_ + tid];
    g_betac[b * C_ + tid] = s;
    if (tid == 0) {
        float sa = 0.f, sm = 0.f;
#pragma unroll
        for (int ch = 0; ch < NCHUNK; ++ch) {
            sa += g_a_part[b * NCHUNK + ch];
            sm += g_m_part[b * NCHUNK + ch];
        }
        g_a[b] = sa; g_m[b] = sm;
    }

    float vals[16];
    float mx = -INFINITY;
#pragma unroll
    for (int i = 0; i < 16; ++i) {
        vals[i] = g_mask[(size_t)b * HW + tid + i * 256];
        mx = fmaxf(mx, vals[i]);
    }
    mx = waveMax(mx);
    if (lane == 0) sh[wave] = mx;
    __syncthreads();
    if (tid == 0) { float m2 = sh[0]; for (int w = 1; w < 8; ++w) m2 = fmaxf(m2, sh[w]); sh[0] = m2; }
    __syncthreads();
    mx = sh[0];
    __syncthreads();

    float ds = 0.f;
#pragma unroll
    for (int i = 0; i < 16; ++i) { vals[i] = expf(vals[i] - mx); ds += vals[i]; }
    ds = waveSum(ds);
    if (lane == 0) sh[wave] = ds;
    __syncthreads();
    if (tid == 0) { float t = 0.f; for (int w = 0; w < 8; ++w) t += sh[w]; sh[0] = t; }
    __syncthreads();
    float inv = 1.0f / sh[0];
#pragma unroll
    for (int i = 0; i < 16; ++i)
        g_mask[(size_t)b * HW + tid + i * 256] = vals[i] * inv;
}

// ---------------------------------------------------------------------------
// Pass 2 over x (float4): context[b,c] = sum_s x[b,c,s] * weight[b,s]
// Wave handles 4 channels; weight loads amortized across the 4.
// ---------------------------------------------------------------------------
__global__ void context_kernel(const float* __restrict__ x,
                               const float* __restrict__ g_w,
                               float* __restrict__ g_ctx) {
    int b  = blockIdx.x >> 3;
    int cg = blockIdx.x & 7;
    int wave = threadIdx.x >> 5, lane = threadIdx.x & 31;
    int cbase = cg * 32 + wave * 4;
    const float4* xb4 = (const float4*)(x + (size_t)b * C_ * HW);
    const float4* wb4 = (const float4*)(g_w + (size_t)b * HW);

    float a0 = 0.f, a1 = 0.f, a2 = 0.f, a3 = 0.f;
#pragma unroll 2
    for (int it = 0; it < 32; ++it) {
        int s4 = lane + it * 32;          // float4 index into 1024
        float4 w = wb4[s4];
        float4 v0 = xb4[(size_t)(cbase + 0) * HW4 + s4];
        float4 v1 = xb4[(size_t)(cbase + 1) * HW4 + s4];
        float4 v2 = xb4[(size_t)(cbase + 2) * HW4 + s4];
        float4 v3 = xb4[(size_t)(cbase + 3) * HW4 + s4];
        a0 = fmaf(v0.x, w.x, fmaf(v0.y, w.y, fmaf(v0.z, w.z, fmaf(v0.w, w.w, a0))));
        a1 = fmaf(v1.x, w.x, fmaf(v1.y, w.y, fmaf(v1.z, w.z, fmaf(v1.w, w.w, a1))));
        a2 = fmaf(v2.x, w.x, fmaf(v2.y, w.y, fmaf(v2.z, w.z, fmaf(v2.w, w.w, a2))));
        a3 = fmaf(v3.x, w.x, fmaf(v3.y, w.y, fmaf(v3.z, w.z, fmaf(v3.w, w.w, a3))));
    }
    a0 = waveSum(a0); a1 = waveSum(a1); a2 = waveSum(a2); a3 = waveSum(a3);
    if (lane == 0) {
        g_ctx[b * C_ + cbase + 0] = a0;
        g_ctx[b * C_ + cbase + 1] = a1;
        g_ctx[b * C_ + cbase + 2] = a2;
        g_ctx[b * C_ + cbase + 3] = a3;
    }
}

// ---------------------------------------------------------------------------
// GEMM1 (WMMA f32 16x16x4): t = context[64,256] @ w_cm1^T[256,128] + b_cm1
// A frag: lanes 0-15 hold (M=lane, K=k0..k0+1), lanes 16-31 (M=lane-16, K=k0+2..3)
// B frag mirrored; D: VGPR r -> M=r (+8 hi lanes), N=lane&15.
// ---------------------------------------------------------------------------
__global__ void gemm1_wmma_kernel(const float* __restrict__ g_ctx,
                                  const float* __restrict__ w_cm1,
                                  const float* __restrict__ b_cm1,
                                  float* __restrict__ g_t) {
    int wave = threadIdx.x >> 5, lane = threadIdx.x & 31;
    int r16 = lane & 15;
    int hi  = (lane >= 16) ? 1 : 0;
    int nt  = wave;                 // 8 waves x 16 cols = 128 cols
    int n   = nt * 16 + r16;
    for (int mt = 0; mt < 4; ++mt) {
        int m = mt * 16 + r16;
        v8f acc = {};
#pragma unroll 4
        for (int k0 = 0; k0 < C_; k0 += 4) {
            int kA = k0 + 2 * hi;
            v2f a, bm;
            a.x  = g_ctx[m * C_ + kA];     a.y  = g_ctx[m * C_ + kA + 1];
            bm.x = w_cm1[n * C_ + kA];     bm.y = w_cm1[n * C_ + kA + 1];
            acc = __builtin_amdgcn_wmma_f32_16x16x4_f32(
                false, a, false, bm, (short)0, acc, false, false);
        }
        float bias = b_cm1[n];
#pragma unroll
        for (int r = 0; r < 8; ++r) {
            int row = mt * 16 + r + hi * 8;
            g_t[row * PLANES + n] = acc[r] + bias;
        }
    }
}

// ---------------------------------------------------------------------------
// LayerNorm over PLANES per row, then exact GELU.
// ---------------------------------------------------------------------------
__global__ void ln_gelu_kernel(float* __restrict__ g_t,
                               const float* __restrict__ ln_w,
                               const float* __restrict__ ln_b) {
    int row = blockIdx.x, tid = threadIdx.x;   // 128 threads = 4 waves
    int wave = tid >> 5, lane = tid & 31;
    __shared__ float sh[4];
    float v = g_t[row * PLANES + tid];
    float s = waveSum(v);
    if (lane == 0) sh[wave] = s;
    __syncthreads();
    if (tid == 0) sh[0] = (sh[0] + sh[1] + sh[2] + sh[3]) * (1.0f / PLANES);
    __syncthreads();
    float mu = sh[0];
    __syncthreads();
    float d = v - mu;
    float s2 = waveSum(d * d);
    if (lane == 0) sh[wave] = s2;
    __syncthreads();
    if (tid == 0) sh[0] = (sh[0] + sh[1] + sh[2] + sh[3]) * (1.0f / PLANES);
    __syncthreads();
    float var = sh[0];
    float t = d * rsqrtf(var + EPSF) * ln_w[tid] + ln_b[tid];
    t = 0.5f * t * (1.0f + erff(t * 0.70710678118654752f));
    g_t[row * PLANES + tid] = t;
}

// ---------------------------------------------------------------------------
// GEMM2 (WMMA f32): beta_g = t[64,128] @ w_cm2^T[128,256], fused epilogue
// out_vec = beta_g + b_cm2 + beta_c + beta_s
// ---------------------------------------------------------------------------
__global__ void gemm2_wmma_kernel(const float* __restrict__ g_t,
                                  const float* __restrict__ w_cm2,
                                  const float* __restrict__ b_cm2,
                                  const float* __restrict__ g_betac,
                                  const float* __restrict__ g_a,
                                  const float* __restrict__ g_m,
                                  float* __restrict__ g_outvec) {
    int wave = threadIdx.x >> 5, lane = threadIdx.x & 31;
    int r16 = lane & 15;
    int hi  = (lane >= 16) ? 1 : 0;
    const float inv_hw = 1.0f / 4096.0f;
    for (int half = 0; half < 2; ++half) {
        int nt = wave + half * 8;   // 16 N-tiles of 16 = 256 cols
        int n  = nt * 16 + r16;
        for (int mt = 0; mt < 4; ++mt) {
            int m = mt * 16 + r16;
            v8f acc = {};
#pragma unroll 4
            for (int k0 = 0; k0 < PLANES; k0 += 4) {
                int kA = k0 + 2 * hi;
                v2f a, bm;
                a.x  = g_t[m * PLANES + kA];      a.y  = g_t[m * PLANES + kA + 1];
                bm.x = w_cm2[n * PLANES + kA];    bm.y = w_cm2[n * PLANES + kA + 1];
                acc = __builtin_amdgcn_wmma_f32_16x16x4_f32(
                    false, a, false, bm, (short)0, acc, false, false);
            }
            float bias = b_cm2[n];
#pragma unroll
            for (int r = 0; r < 8; ++r) {
                int row = mt * 16 + r + hi * 8;
                float beta_s = ((n & 1) ? g_m[row] : g_a[row]) * inv_hw;
                float beta_c = g_betac[row * C_ + n] * inv_hw;
                g_outvec[row * C_ + n] = acc[r] + bias + beta_c + beta_s;
            }
        }
    }
}

// ---------------------------------------------------------------------------
// Head: net (256->16 relu ->8), dynamic kernel attention, softmax(out/30).
// One thread per batch row (tiny).
// ---------------------------------------------------------------------------
__global__ void head_kernel(const float* __restrict__ g_outvec,
                            const float* __restrict__ w_net1,
                            const float* __restrict__ w_net2,
                            const float* __restrict__ w_fc,
                            const float* __restrict__ bn_w,
                            const float* __restrict__ bn_b,
                            const float* __restrict__ bn_mean,
                            const float* __restrict__ bn_var,
                            const float* __restrict__ w_kfc,
                            float* __restrict__ out) {
    int b = threadIdx.x;
    if (b >= B_) return;
    const float* ov = g_outvec + b * C_;
    float h[HIDDEN];
#pragma unroll
    for (int j = 0; j < HIDDEN; ++j) {
        float s = 0.f;
        for (int c = 0; c < C_; ++c) s = fmaf(ov[c], w_net1[j * C_ + c], s);
        h[j] = fmaxf(s, 0.f);
    }
    float o[K_];
#pragma unroll
    for (int k = 0; k < K_; ++k) {
        float s = 0.f;
#pragma unroll
        for (int j = 0; j < HIDDEN; ++j) s = fmaf(h[j], w_net2[k * HIDDEN + j], s);
        o[k] = s;
    }
    float ka[ATT];
#pragma unroll
    for (int a = 0; a < ATT; ++a) {
        float s = 0.f;
#pragma unroll
        for (int k = 0; k < K_; ++k) s = fmaf(o[k], w_fc[a * K_ + k], s);
        s = (s - bn_mean[a]) * rsqrtf(bn_var[a] + EPSF) * bn_w[a] + bn_b[a];
        ka[a] = fmaxf(s, 0.f);
    }
    float t[K_];
    float mx = -INFINITY;
#pragma unroll
    for (int k = 0; k < K_; ++k) {
        float s = 0.f;
#pragma unroll
        for (int a = 0; a < ATT; ++a) s = fmaf(ka[a], w_kfc[k * ATT + a], s);
        float kern = 1.0f / (1.0f + expf(-s));
        t[k] = o[k] * kern * (1.0f / TEMPF);
        mx = fmaxf(mx, t[k]);
    }
    float den = 0.f;
#pragma unroll
    for (int k = 0; k < K_; ++k) { t[k] = expf(t[k] - mx); den += t[k]; }
    float inv = 1.0f / den;
#pragma unroll
    for (int k = 0; k < K_; ++k) out[b * K_ + k] = t[k] * inv;
}

extern "C" void kernel_launch(void* const* d_in, const int* in_sizes, int n_in,
                              void* d_out, int out_size, void* d_ws, size_t ws_size,
                              hipStream_t stream) {
    const float* x       = (const float*)d_in[0];
    const float* w_mask  = (const float*)d_in[1];
    const float* b_mask  = (const float*)d_in[2];
    const float* w_cm1   = (const float*)d_in[3];
    const float* b_cm1   = (const float*)d_in[4];
    const float* ln_w    = (const float*)d_in[5];
    const float* ln_b    = (const float*)d_in[6];
    const float* w_cm2   = (const float*)d_in[7];
    const float* b_cm2   = (const float*)d_in[8];
    const float* w_net1  = (const float*)d_in[9];
    const float* w_net2  = (const float*)d_in[10];
    const float* w_fc    = (const float*)d_in[11];
    const float* bn_w    = (const float*)d_in[12];
    const float* bn_b    = (const float*)d_in[13];
    const float* bn_mean = (const float*)d_in[14];
    const float* bn_var  = (const float*)d_in[15];
    const float* w_kfc   = (const float*)d_in[16];
    float* out = (float*)d_out;

    float* ws        = (float*)d_ws;
    float* g_betac   = ws;               // 16384
    float* g_a       = ws + 16384;       // 64
    float* g_m       = ws + 16448;       // 64
    float* g_mask    = ws + 16512;       // 262144 (logits -> softmax weights)
    float* g_ctx     = ws + 278656;      // 16384
    float* g_t       = ws + 295040;      // 8192
    float* g_outvec  = ws + 303232;      // 16384
    float* g_bc_part = ws + 319616;      // 65536
    float* g_a_part  = ws + 385152;      // 256
    float* g_m_part  = ws + 385408;      // 256  -> total ~1.54 MB

    pass1_kernel<<<dim3(B_ * NCHUNK), dim3(256), 0, stream>>>(
        x, w_mask, b_mask, g_mask, g_bc_part, g_a_part, g_m_part);
    reduce_softmax_kernel<<<dim3(B_), dim3(256), 0, stream>>>(
        g_mask, g_bc_part, g_a_part, g_m_part, g_betac, g_a, g_m);
    context_kernel<<<dim3(B_ * 8), dim3(256), 0, stream>>>(x, g_mask, g_ctx);
    gemm1_wmma_kernel<<<dim3(1), dim3(256), 0, stream>>>(g_ctx, w_cm1, b_cm1, g_t);
    ln_gelu_kernel<<<dim3(B_), dim3(128), 0, stream>>>(g_t, ln_w, ln_b);
    gemm2_wmma_kernel<<<dim3(1), dim3(256), 0, stream>>>(
        g_t, w_cm2, b_cm2, g_betac, g_a, g_m, g_outvec);
    head_kernel<<<dim3(1), dim3(64), 0, stream>>>(
        g_outvec, w_net1, w_net2, w_fc, bn_w, bn_b, bn_mean, bn_var, w_kfc, out);
}